// MiniRocketTransform_27084063769233
// MI455X (gfx1250) — compile-verified
//
#include <hip/hip_runtime.h>
#include <hip/hip_bf16.h>

typedef __attribute__((ext_vector_type(2))) float v2f;
typedef __attribute__((ext_vector_type(8))) float v8f;

#define IN_LEN   4096
#define KLEN     9
#define NDIL     6          // dilations 1..6 (from np.logspace(0, log10(511), 84, base=2) -> unique ints)
#define NFEAT    10000
#define MAXK     96         // padded kernel rows (6 tiles of 16)
#define TAPS_PAD 12         // 9 taps zero-padded to 3 chunks of K=4

// ---------------- kernel 0: zero the bias accumulators (re-run every call) ----
__global__ void zero_bias_k(float* __restrict__ bias) {
    if (threadIdx.x < 3) bias[threadIdx.x] = 0.0f;
}

// ---------------- kernel 1: conv0 (kernel 0, dilation 1) + per-sample quantiles
// one block per batch sample; bitonic sort of 4088 conv outputs (padded to 4096)
__global__ __launch_bounds__(256) void bias_k(const float* __restrict__ x,
                                              const float* __restrict__ kers,
                                              float* __restrict__ bias,
                                              float invB) {
    __shared__ float s[IN_LEN];
    const int b   = blockIdx.x;
    const int tid = threadIdx.x;
    const float* xb = x + (size_t)b * IN_LEN;

    float w0[KLEN];
#pragma unroll
    for (int j = 0; j < KLEN; ++j) w0[j] = kers[j];

    const int L0 = IN_LEN - (KLEN - 1);   // 4088 valid positions
    for (int t = tid; t < IN_LEN; t += 256) {
        if (t < L0) {
            float c = 0.0f;
#pragma unroll
            for (int j = 0; j < KLEN; ++j) c += w0[j] * xb[t + j];
            s[t] = c;
        } else {
            s[t] = 1e30f;   // pad sorts to the top; never selected
        }
    }

    // bitonic sort, ascending
    for (int k = 2; k <= IN_LEN; k <<= 1) {
        for (int j = k >> 1; j > 0; j >>= 1) {
            __syncthreads();
            for (int i = tid; i < IN_LEN; i += 256) {
                int ixj = i ^ j;
                if (ixj > i) {
                    float a = s[i], c = s[ixj];
                    bool up = (i & k) == 0;
                    if ((a > c) == up) { s[i] = c; s[ixj] = a; }
                }
            }
        }
    }
    __syncthreads();

    if (tid == 0) {
        // n = 4088: positions q*(n-1) = 1021.75, 2043.5, 3065.25 (linear interp)
        float q25 = s[1021] * 0.25f + s[1022] * 0.75f;
        float q50 = (s[2043] + s[2044]) * 0.5f;
        float q75 = s[3065] * 0.75f + s[3066] * 0.25f;
        atomicAdd(&bias[0], q25 * invB);
        atomicAdd(&bias[1], q50 * invB);
        atomicAdd(&bias[2], q75 * invB);
    }
}

// ---------------- kernel 2: all dilated convs via WMMA f32 16x16x4 + PPV + L2 norm
// one block (8 waves) per batch sample; waves split (dilation x kernel-tile) pairs
__global__ __launch_bounds__(256) void feat_k(const float* __restrict__ x,
                                              const float* __restrict__ kers,
                                              const float* __restrict__ bias,
                                              float* __restrict__ out,
                                              int K, int KT) {
    __shared__ float xs[IN_LEN + 64];            // +64 zeros: safe reads for partial tiles
    __shared__ float wk[MAXK * TAPS_PAD];        // kernels padded (rows>=K and taps>=9 are 0)
    __shared__ float feats[MAXK * NDIL * 3];
    __shared__ float red[8];

    const int b    = blockIdx.x;
    const int tid  = threadIdx.x;
    const int wave = tid >> 5;
    const int lane = tid & 31;
    const int lh   = lane & 15;
    const bool hi  = lane >= 16;

    const float* xb = x + (size_t)b * IN_LEN;
    for (int i = tid; i < IN_LEN; i += 256) xs[i] = xb[i];
    for (int i = IN_LEN + tid; i < IN_LEN + 64; i += 256) xs[i] = 0.0f;
    for (int i = tid; i < MAXK * TAPS_PAD; i += 256) {
        int r = i / TAPS_PAD, c = i - r * TAPS_PAD;
        wk[i] = (r < K && c < KLEN) ? kers[r * KLEN + c] : 0.0f;
    }
    for (int i = tid; i < MAXK * NDIL * 3; i += 256) feats[i] = 0.0f;
    __syncthreads();

    const float b0 = bias[0], b1 = bias[1], b2 = bias[2];

    const int P = NDIL * KT;                     // (dilation, kernel-tile) pairs
    for (int p = wave; p < P; p += 8) {          // wave-uniform loop: EXEC stays all-1s
        const int dIdx = p / KT;
        const int kt   = p - dIdx * KT;
        const int d    = dIdx + 1;               // dilations 1..6
        const int Ld   = IN_LEN - (KLEN - 1) * d;
        const int nT   = (Ld + 15) >> 4;

        // A fragments: 16 kernels x 4 taps per chunk (f32 A 16x4 layout:
        // lanes 0-15 hold taps {4c,4c+1}, lanes 16-31 taps {4c+2,4c+3})
        const int m = kt * 16 + lh;
        v2f a[3];
#pragma unroll
        for (int c = 0; c < 3; ++c) {
            int tap = 4 * c + (hi ? 2 : 0);
            a[c].x = wk[m * TAPS_PAD + tap];
            a[c].y = wk[m * TAPS_PAD + tap + 1];
        }

        int acc[8][3];
#pragma unroll
        for (int v = 0; v < 8; ++v)
#pragma unroll
            for (int q = 0; q < 3; ++q) acc[v][q] = 0;

        for (int tt = 0; tt < nT; ++tt) {
            const int t = (tt << 4) + lh;        // time column for this lane
            v8f dfr = {0.f,0.f,0.f,0.f,0.f,0.f,0.f,0.f};
#pragma unroll
            for (int c = 0; c < 3; ++c) {
                int tap = 4 * c + (hi ? 2 : 0);
                v2f bf;
                bf.x = xs[t + tap * d];
                bf.y = xs[t + (tap + 1) * d];
                // D = A(16x4) * B(4x16) + C, chained over the 3 tap-chunks
                dfr = __builtin_amdgcn_wmma_f32_16x16x4_f32(
                        false, a[c], false, bf, (short)0, dfr, false, false);
            }
            const bool valid = t < Ld;
#pragma unroll
            for (int v = 0; v < 8; ++v) {
                float cv = dfr[v];
                if (valid) {
                    acc[v][0] += (cv > b0);
                    acc[v][1] += (cv > b1);
                    acc[v][2] += (cv > b2);
                }
            }
        }

        // reduce counts over each 16-lane group (rows v in lanes 0-15, v+8 in 16-31)
#pragma unroll
        for (int v = 0; v < 8; ++v) {
#pragma unroll
            for (int q = 0; q < 3; ++q) {
                int sv = acc[v][q];
                sv += __shfl_xor(sv, 1);
                sv += __shfl_xor(sv, 2);
                sv += __shfl_xor(sv, 4);
                sv += __shfl_xor(sv, 8);
                acc[v][q] = sv;
            }
        }

        if (lh == 0) {
            const float invL = 1.0f / (float)Ld;
#pragma unroll
            for (int v = 0; v < 8; ++v) {
                int row = kt * 16 + v + (hi ? 8 : 0);
                if (row < K) {
                    int base = (row * NDIL + dIdx) * 3;   // reshape (K, D, 3)
                    feats[base + 0] = acc[v][0] * invL;
                    feats[base + 1] = acc[v][1] * invL;
                    feats[base + 2] = acc[v][2] * invL;
                }
            }
        }
    }
    __syncthreads();

    // L2 norm over F = K*18 real features; write normalized vector, zero tail
    const int F = K * NDIL * 3;
    float ss = 0.0f;
    for (int i = tid; i < F; i += 256) { float f = feats[i]; ss += f * f; }
    ss += __shfl_xor(ss, 1);
    ss += __shfl_xor(ss, 2);
    ss += __shfl_xor(ss, 4);
    ss += __shfl_xor(ss, 8);
    ss += __shfl_xor(ss, 16);
    if (lane == 0) red[wave] = ss;
    __syncthreads();
    if (tid == 0) {
        float t = 0.0f;
        for (int w = 0; w < 8; ++w) t += red[w];
        red[0] = t;
    }
    __syncthreads();
    const float inv = 1.0f / fmaxf(sqrtf(red[0]), 1e-12f);
    float* ob = out + (size_t)b * NFEAT;
    for (int i = tid; i < NFEAT; i += 256)
        ob[i] = (i < F) ? feats[i] * inv : 0.0f;
}

extern "C" void kernel_launch(void* const* d_in, const int* in_sizes, int n_in,
                              void* d_out, int out_size, void* d_ws, size_t ws_size,
                              hipStream_t stream) {
    const float* x    = (const float*)d_in[0];
    const float* kers = (const float*)d_in[1];
    float* out     = (float*)d_out;
    float* biasAcc = (float*)d_ws;   // 3 floats of scratch

    const int B = in_sizes[0] / IN_LEN;
    int K = in_sizes[1] / KLEN;
    if (K > MAXK) K = MAXK;
    const int KT = (K + 15) / 16;

    zero_bias_k<<<1, 32, 0, stream>>>(biasAcc);
    bias_k<<<B, 256, 0, stream>>>(x, kers, biasAcc, 1.0f / (float)B);
    feat_k<<<B, 256, 0, stream>>>(x, kers, biasAcc, out, K, KT);
}